// SpikingActivation_57595511439952
// MI455X (gfx1250) — compile-verified
//
#include <hip/hip_runtime.h>
#include <hip/hip_bf16.h>

typedef __attribute__((ext_vector_type(2))) float v2f;
typedef __attribute__((ext_vector_type(8))) float v8f;

#define SA_DT     0.001f
#define SA_INVDT  1000.0f
#define SA_T      128
#define SA_N      4096
#define SA_B      64

// One wave (32 lanes) owns one (batch, 16-neuron tile) column and performs the
// full T=128 cumsum as a blocked lower-triangular matmul on the FP32 WMMA pipe:
//   S = L(128x128, lower-tri ones) @ X(128x16),  X = relu(in)*dt
// Per 16-step time tile: 4x V_WMMA_F32_16X16X4_F32 with triangular A gives the
// in-tile inclusive cumsum; 4x with all-ones A gives the column-sum tile that
// carries into every later tile (plain vector adds). FP32 throughout so the
// floor() boundaries match the reference. Memory traffic is exactly 1R+1W per
// element -> bandwidth-roofline (~11.5 us at 23.3 TB/s).
__global__ __launch_bounds__(256) void spiking_cumsum_wmma_kernel(
    const float* __restrict__ in,   // (B, T, N)
    const float* __restrict__ v0,   // (B, N)
    float* __restrict__ out)        // (B, T, N)
{
    const int lane = threadIdx.x & 31;
    const int wave = threadIdx.x >> 5;
    const int b    = blockIdx.x >> 5;        // 32 blocks span N per batch
    const int ng   = blockIdx.x & 31;
    const int ntile = ng * 8 + wave;         // 8 waves/block * 16 cols = 128 cols/block
    const int ncol  = lane & 15;             // column within tile (N index, = lane%16)
    const int half  = lane >> 4;             // 0: rows/K {0..} low, 1: high half
    const int n     = ntile * 16 + ncol;

    const float* inb  = in  + (size_t)b * SA_T * SA_N + n;
    float*       outb = out + (size_t)b * SA_T * SA_N + n;

    // ---- A operands (constants, built per lane) ----
    // A is 16x4 fp32: lane m=lane&15, VGPR j holds K = j + 2*half.
    const int m = lane & 15;
    v2f onesA;  onesA.x = 1.0f;  onesA.y = 1.0f;
    v2f triA[4];
#pragma unroll
    for (int c = 0; c < 4; ++c) {
        const int k0 = c * 4 + (half ? 2 : 0);
        triA[c].x = (k0     <= m) ? 1.0f : 0.0f;   // L16[m,k] = (k <= m)
        triA[c].y = (k0 + 1 <= m) ? 1.0f : 0.0f;
    }

    // ---- accumulate S tiles ----
    v8f acc[8] = {};                 // 8 output time-tiles, 16x16 f32 each
#pragma unroll
    for (int kb = 0; kb < 8; ++kb) {
        if (kb < 7) {
            // streaming hint for the next time tile
            __builtin_prefetch(inb + (size_t)(kb + 1) * 16 * SA_N, 0, 0);
        }
        v8f ones_s = {};             // colsum(X_kb) broadcast down all 16 rows
#pragma unroll
        for (int c = 0; c < 4; ++c) {
            // B is 4x16 fp32: VGPR j, this lane holds (K = j + 2*half, N = ncol)
            const int t0 = kb * 16 + c * 4 + (half ? 2 : 0);
            const float x0 = inb[(size_t)t0 * SA_N];
            const float x1 = inb[(size_t)(t0 + 1) * SA_N];
            v2f Bc;
            Bc.x = fmaxf(x0, 0.0f) * SA_DT;
            Bc.y = fmaxf(x1, 0.0f) * SA_DT;
            acc[kb] = __builtin_amdgcn_wmma_f32_16x16x4_f32(
                false, triA[c], false, Bc, (short)0, acc[kb], false, false);
            ones_s  = __builtin_amdgcn_wmma_f32_16x16x4_f32(
                false, onesA,  false, Bc, (short)0, ones_s,  false, false);
        }
#pragma unroll
        for (int i = kb + 1; i < 8; ++i)
            acc[i] += ones_s;        // carry of this block into all later tiles
    }

    // ---- floor / diff / store ----
    // C/D layout: VGPR r holds row M=r (lanes 0-15) and M=r+8 (lanes 16-31),
    // column N = lane%16. So each lane's 8 components are 8 consecutive time
    // steps; only the row7->row8 and tile-boundary carries cross lane halves.
    const float v0v  = v0[(size_t)b * SA_N + n];
    float carry = floorf(v0v);       // floor(S_{-1}); consumed by half0 lanes

#pragma unroll
    for (int i = 0; i < 8; ++i) {
        float f[8];
#pragma unroll
        for (int r = 0; r < 8; ++r)
            f[r] = floorf(acc[i][r] + v0v);
        // half0 receives row15's floor (next tile's carry);
        // half1 receives row7's floor (prev for its row8).
        const float cross = __shfl_xor(f[7], 16, 32);
        const float prev0 = half ? cross : carry;

        const int tb = i * 16 + (half ? 8 : 0);
        outb[(size_t)tb * SA_N] = (f[0] - prev0) * SA_INVDT;
#pragma unroll
        for (int r = 1; r < 8; ++r)
            outb[(size_t)(tb + r) * SA_N] = (f[r] - f[r - 1]) * SA_INVDT;

        if (!half) carry = cross;    // row15 floor becomes next tile's carry
    }
}

extern "C" void kernel_launch(void* const* d_in, const int* in_sizes, int n_in,
                              void* d_out, int out_size, void* d_ws, size_t ws_size,
                              hipStream_t stream) {
    const float* in  = (const float*)d_in[0];   // (64, 128, 4096) f32
    const float* v0  = (const float*)d_in[1];   // (64, 4096) f32
    float*       out = (float*)d_out;           // (64, 128, 4096) f32
    (void)in_sizes; (void)n_in; (void)out_size; (void)d_ws; (void)ws_size;

    // 64 batches * 32 blocks; each block = 8 waves = 8 neuron tiles (128 cols)
    dim3 grid(SA_B * 32);
    dim3 block(256);
    spiking_cumsum_wmma_kernel<<<grid, block, 0, stream>>>(in, v0, out);
}